// Sampler_33938831573202
// MI455X (gfx1250) — compile-verified
//
#include <hip/hip_runtime.h>
#include <hip/hip_bf16.h>

// ---------------------------------------------------------------------------
// Problem constants (from the reference)
// ---------------------------------------------------------------------------
#define NS 4096      // support rows
#define NQ 4096      // query rows
#define NW 64        // ways
#define KSHOT 64
#define TOPK 32
#define FDIM 768
#define EPS 1e-3f

typedef __attribute__((ext_vector_type(2))) float v2f;
typedef __attribute__((ext_vector_type(8))) float v8f;
typedef int v4i __attribute__((vector_size(4 * sizeof(int))));

#define AS1 __attribute__((address_space(1)))
#define AS3 __attribute__((address_space(3)))

// ---------------------------------------------------------------------------
// CDNA5 async global->LDS copy (ASYNCcnt-tracked, no staging VGPRs).
// Builtin signature (from hipcc diagnostic): param0 = v4i __device__(AS1)*.
// ---------------------------------------------------------------------------
__device__ __forceinline__ void async_copy16(const float* gsrc, float* ldst) {
#if __has_builtin(__builtin_amdgcn_global_load_async_to_lds_b128)
    __builtin_amdgcn_global_load_async_to_lds_b128(
        (AS1 v4i*)(uintptr_t)gsrc,
        (AS3 v4i*)(AS3 float*)ldst,
        0, 0);
#else
    const unsigned loff = (unsigned)(uintptr_t)(AS3 float*)ldst;
    asm volatile("global_load_async_to_lds_b128 %0, %1, off"
                 :: "v"(loff), "v"((unsigned long long)(uintptr_t)gsrc)
                 : "memory");
#endif
}

#if __has_builtin(__builtin_amdgcn_s_wait_asynccnt)
#define WAIT_ASYNC(n) __builtin_amdgcn_s_wait_asynccnt(n)
#else
#define WAIT_ASYNC(n) asm volatile("s_wait_asynccnt " #n ::: "memory")
#endif

// ---------------------------------------------------------------------------
// Kernel 0: zero the accuracy counter (harness does not re-init d_ws)
// ---------------------------------------------------------------------------
__global__ void init_kernel(int* counter) {
    if (threadIdx.x == 0) *counter = 0;
}

// ---------------------------------------------------------------------------
// Kernel 1: sq_s[s] = ||support_s||^2   (one 256-thread block per row)
// ---------------------------------------------------------------------------
__global__ __launch_bounds__(256) void sqs_kernel(const float* __restrict__ S,
                                                  float* __restrict__ sqs) {
    __shared__ float red[256];
    const int s = blockIdx.x, t = threadIdx.x;
    const float* row = S + (size_t)s * FDIM;
    float a = row[t], b = row[t + 256], c = row[t + 512];
    float v = a * a + b * b + c * c;
    red[t] = v;
    __syncthreads();
    for (int s2 = 128; s2 > 0; s2 >>= 1) {
        if (t < s2) red[t] += red[t + s2];
        __syncthreads();
    }
    if (t == 0) sqs[s] = red[0];
}

// ---------------------------------------------------------------------------
// Kernel 2: class softmax + repeated-entropy divisor
//   clsdiv[r][w] = softmax(cls[r])[w] / (1 + max(0, -64 * sum_w p*log(p+eps)))
// ---------------------------------------------------------------------------
__global__ __launch_bounds__(64) void cls_kernel(const float* __restrict__ cls,
                                                 float* __restrict__ clsdiv) {
    __shared__ float red[64];
    const int r = blockIdx.x, t = threadIdx.x;
    float x = cls[(size_t)r * NW + t];

    red[t] = x;
    __syncthreads();
    for (int s2 = 32; s2 > 0; s2 >>= 1) {
        if (t < s2) red[t] = fmaxf(red[t], red[t + s2]);
        __syncthreads();
    }
    float m = red[0];
    __syncthreads();

    float e = expf(x - m);
    red[t] = e;
    __syncthreads();
    for (int s2 = 32; s2 > 0; s2 >>= 1) {
        if (t < s2) red[t] += red[t + s2];
        __syncthreads();
    }
    float Z = red[0];
    __syncthreads();

    float p = e / Z;
    red[t] = p * logf(p + EPS);
    __syncthreads();
    for (int s2 = 32; s2 > 0; s2 >>= 1) {
        if (t < s2) red[t] += red[t + s2];
        __syncthreads();
    }
    float H = fmaxf(0.0f, -64.0f * red[0]);
    clsdiv[(size_t)r * NW + t] = p / (1.0f + H);
}

// ---------------------------------------------------------------------------
// Kernel 3: WMMA fp32 GEMM with double-buffered async global->LDS staging.
//   L[q][s] = 2 * dot(query_q, support_s) - ||support_s||^2
// Block tile 128(q) x 128(s), K chunk 32; 8 waves x (64q x 32s) each.
// Async pipeline: issue chunk kc+1 into buf^1, s_wait_asynccnt 8 (in-order
// completion => chunk kc's 8 ops landed), barrier, WMMA from buf.
// ---------------------------------------------------------------------------
#define LDK 36
#define NCHUNK (FDIM / 32)

__device__ __forceinline__ void issue_tile(const float* __restrict__ Q,
                                           const float* __restrict__ S,
                                           int qb, int sb, int k0,
                                           float* lq, float* ls, int t) {
#pragma unroll
    for (int j = 0; j < 4; ++j) {
        int idx = t + (j << 8);          // 0..1023
        int row = idx >> 3;
        int kq = (idx & 7) << 2;
        async_copy16(Q + (size_t)(qb + row) * FDIM + k0 + kq, lq + row * LDK + kq);
        async_copy16(S + (size_t)(sb + row) * FDIM + k0 + kq, ls + row * LDK + kq);
    }
}

__global__ __launch_bounds__(256) void gemm_kernel(const float* __restrict__ Q,
                                                   const float* __restrict__ S,
                                                   const float* __restrict__ sqs,
                                                   float* __restrict__ L) {
    __shared__ float lQ[2][128 * LDK];
    __shared__ float lS[2][128 * LDK];

    const int t = threadIdx.x;
    const int qb = blockIdx.y * 128;
    const int sb = blockIdx.x * 128;
    const int lane = t & 31, wid = t >> 5;
    const int half = lane >> 4, l16 = lane & 15;
    const int wq = (wid & 1) * 64;   // wave q offset within block tile
    const int ws = (wid >> 1) * 32;  // wave s offset within block tile

    v8f zero = {0.f, 0.f, 0.f, 0.f, 0.f, 0.f, 0.f, 0.f};
    v8f acc[4][2];
#pragma unroll
    for (int mi = 0; mi < 4; ++mi)
#pragma unroll
        for (int ni = 0; ni < 2; ++ni) acc[mi][ni] = zero;

    // Prologue: chunk 0 in flight.
    issue_tile(Q, S, qb, sb, 0, lQ[0], lS[0], t);

    for (int kc = 0; kc < NCHUNK; ++kc) {
        const int b = kc & 1;
        if (kc + 1 < NCHUNK) {
            issue_tile(Q, S, qb, sb, (kc + 1) * 32, lQ[b ^ 1], lS[b ^ 1], t);
            WAIT_ASYNC(8);   // keep only the 8 newest (next chunk) outstanding
        } else {
            WAIT_ASYNC(0);
        }
        __syncthreads();     // all waves' chunk-kc data visible in LDS

        const float* lq = lQ[b];
        const float* ls = lS[b];
#pragma unroll
        for (int ks = 0; ks < 8; ++ks) {
            const int koff = (ks << 2) + (half << 1);
            v2f aA[4], bB[2];
#pragma unroll
            for (int mi = 0; mi < 4; ++mi)
                aA[mi] = *(const v2f*)(lq + (wq + mi * 16 + l16) * LDK + koff);
#pragma unroll
            for (int ni = 0; ni < 2; ++ni)
                bB[ni] = *(const v2f*)(ls + (ws + ni * 16 + l16) * LDK + koff);
#pragma unroll
            for (int mi = 0; mi < 4; ++mi)
#pragma unroll
                for (int ni = 0; ni < 2; ++ni)
                    acc[mi][ni] = __builtin_amdgcn_wmma_f32_16x16x4_f32(
                        false, aA[mi], false, bB[ni], (short)0, acc[mi][ni],
                        false, false);
        }
        __syncthreads();     // protect buf b before it is refilled at kc+2
    }

    // Epilogue: D layout -> M(q) = v + 8*half, N(s) = lane%16.
    // Half-wave lanes hit 16 consecutive s -> coalesced 64B stores.
#pragma unroll
    for (int ni = 0; ni < 2; ++ni) {
        const int s = sb + ws + ni * 16 + l16;
        const float sq = sqs[s];
#pragma unroll
        for (int mi = 0; mi < 4; ++mi) {
#pragma unroll
            for (int v = 0; v < 8; ++v) {
                const int q = qb + wq + mi * 16 + v + (half << 3);
                L[(size_t)q * NS + s] = 2.0f * acc[mi][ni][v] - sq;
            }
        }
    }
}

// ---------------------------------------------------------------------------
// Kernel 4: per-query softmax stats over the 4096 supports.
//   m[q] = max_s L[q][s];  Z = sum exp(L-m);  H = max(0, -sum p*log(p+eps))
//   inv[q] = 1 / (Z * (1 + H))  so  sim term = exp(L-m) * inv
// ---------------------------------------------------------------------------
__global__ __launch_bounds__(256) void softstat_kernel(const float* __restrict__ L,
                                                       float* __restrict__ mOut,
                                                       float* __restrict__ invOut) {
    __shared__ float red[256];
    const int q = blockIdx.x, t = threadIdx.x;
    const float4* row4 = (const float4*)(L + (size_t)q * NS);

    float x[16];
#pragma unroll
    for (int i = 0; i < 4; ++i) {
        float4 v = row4[t * 4 + i];
        x[4 * i + 0] = v.x; x[4 * i + 1] = v.y;
        x[4 * i + 2] = v.z; x[4 * i + 3] = v.w;
    }

    float mx = x[0];
#pragma unroll
    for (int i = 1; i < 16; ++i) mx = fmaxf(mx, x[i]);
    red[t] = mx;
    __syncthreads();
    for (int s2 = 128; s2 > 0; s2 >>= 1) {
        if (t < s2) red[t] = fmaxf(red[t], red[t + s2]);
        __syncthreads();
    }
    mx = red[0];
    __syncthreads();

    float e[16], sum = 0.0f;
#pragma unroll
    for (int i = 0; i < 16; ++i) { e[i] = expf(x[i] - mx); sum += e[i]; }
    red[t] = sum;
    __syncthreads();
    for (int s2 = 128; s2 > 0; s2 >>= 1) {
        if (t < s2) red[t] += red[t + s2];
        __syncthreads();
    }
    const float Z = red[0];
    __syncthreads();

    const float invZ = 1.0f / Z;
    float h = 0.0f;
#pragma unroll
    for (int i = 0; i < 16; ++i) {
        float p = e[i] * invZ;
        h += p * logf(p + EPS);
    }
    red[t] = h;
    __syncthreads();
    for (int s2 = 128; s2 > 0; s2 >>= 1) {
        if (t < s2) red[t] += red[t + s2];
        __syncthreads();
    }
    if (t == 0) {
        float H = fmaxf(0.0f, -red[0]);
        mOut[q] = mx;
        invOut[q] = invZ / (1.0f + H);
    }
}

// ---------------------------------------------------------------------------
// Kernel 5: per-support top-32 over the 4096 queries + gather + accuracy.
//   c[q] = clsdiv[q][s>>6] + exp(L[q][s] - m[q]) * inv[q]   (all >= 0)
// 32 rounds of u64 argmax (orderable-float<<12 | (4095-q)) — reproduces
// top_k descending order with smallest-index tie-break.
// ---------------------------------------------------------------------------
__device__ __forceinline__ unsigned orderable(float f) {
    unsigned u = __float_as_uint(f);
    return (u & 0x80000000u) ? ~u : (u | 0x80000000u);
}

__global__ __launch_bounds__(256) void topk_kernel(const float* __restrict__ L,
                                                   const float* __restrict__ clsdiv,
                                                   const float* __restrict__ mRow,
                                                   const float* __restrict__ invRow,
                                                   const float* __restrict__ qemb,
                                                   float* __restrict__ out,
                                                   int* __restrict__ counter) {
    __shared__ unsigned long long sred[256];
    __shared__ int sIdx[TOPK];

    const int s = blockIdx.x, t = threadIdx.x;
    const int way = s >> 6;
    const int lo = way * KSHOT;   // Q_SHOT == KSHOT == 64

    float c[16];
#pragma unroll
    for (int i = 0; i < 16; ++i) {
        const int q = t + (i << 8);
        const float lg = L[(size_t)q * NS + s];
        c[i] = clsdiv[(size_t)q * NW + way] + expf(lg - mRow[q]) * invRow[q];
    }

    int cnt = 0;  // thread-0 accuracy accumulator
    for (int r = 0; r < TOPK; ++r) {
        float bv = c[0];
        int bi = 0;
#pragma unroll
        for (int i = 1; i < 16; ++i)
            if (c[i] > bv) { bv = c[i]; bi = i; }
        const int bq = t + (bi << 8);
        unsigned long long key =
            ((unsigned long long)orderable(bv) << 12) | (unsigned)(4095 - bq);

        sred[t] = key;
        __syncthreads();
        for (int s2 = 128; s2 > 0; s2 >>= 1) {
            if (t < s2 && sred[t + s2] > sred[t]) sred[t] = sred[t + s2];
            __syncthreads();
        }
        const unsigned long long best = sred[0];
        const int wq = 4095 - (int)(best & 0xFFFull);

        if (t == (wq & 255)) {
            const int ci = wq >> 8;
#pragma unroll
            for (int i = 0; i < 16; ++i)
                if (i == ci) c[i] = -1.0f;  // all real values >= 0
        }
        if (t == 0) {
            sIdx[r] = wq;
            if (wq >= lo && wq < lo + KSHOT) ++cnt;
        }
        __syncthreads();
    }

    if (t == 0) atomicAdd(counter, cnt);

    // Gather: 32 rows x 768 floats = 6144 float4, coalesced.
    const float4* q4 = (const float4*)qemb;
    float4* o4 = (float4*)(out + (size_t)s * TOPK * FDIM);
    for (int i = t; i < TOPK * (FDIM / 4); i += 256) {
        const int j = i / (FDIM / 4);
        const int col = i % (FDIM / 4);
        o4[(size_t)j * (FDIM / 4) + col] = q4[(size_t)sIdx[j] * (FDIM / 4) + col];
    }
}

// ---------------------------------------------------------------------------
// Kernel 6: accuracy scalar (deterministic: single int -> float)
// ---------------------------------------------------------------------------
__global__ void fin_kernel(const int* counter, float* accOut) {
    *accOut = (float)(*counter) * (1.0f / (float)(NW * KSHOT * TOPK));
}

// ---------------------------------------------------------------------------
// Launch: workspace layout (bytes):
//   L       : [0,        67108864)   4096*4096 f32 logits
//   clsdiv  : [67108864, 68157440)   4096*64 f32
//   sqs     : [68157440, 68173824)
//   m       : [68173824, 68190208)
//   inv     : [68190208, 68206592)
//   counter : [68206592, 68206596)
// ---------------------------------------------------------------------------
extern "C" void kernel_launch(void* const* d_in, const int* in_sizes, int n_in,
                              void* d_out, int out_size, void* d_ws, size_t ws_size,
                              hipStream_t stream) {
    const float* sup = (const float*)d_in[0];  // [4096,768]
    const float* qry = (const float*)d_in[1];  // [4096,768]
    const float* cls = (const float*)d_in[2];  // [4096,64]
    float* out = (float*)d_out;

    char* ws = (char*)d_ws;
    float* L      = (float*)(ws);
    float* clsdiv = (float*)(ws + 67108864);
    float* sqs    = (float*)(ws + 68157440);
    float* mRow   = (float*)(ws + 68173824);
    float* invRow = (float*)(ws + 68190208);
    int*   counter= (int*)  (ws + 68206592);

    init_kernel<<<1, 1, 0, stream>>>(counter);
    sqs_kernel<<<NS, 256, 0, stream>>>(sup, sqs);
    cls_kernel<<<NS, 64, 0, stream>>>(cls, clsdiv);
    gemm_kernel<<<dim3(NS / 128, NQ / 128), 256, 0, stream>>>(qry, sup, sqs, L);
    softstat_kernel<<<NQ, 256, 0, stream>>>(L, mRow, invRow);
    topk_kernel<<<NS, 256, 0, stream>>>(L, clsdiv, mRow, invRow, qry, out, counter);
    fin_kernel<<<1, 1, 0, stream>>>(counter, out + ((size_t)out_size - 1));
}